// GraphFeatLearningLayer_21217138442472
// MI455X (gfx1250) — compile-verified
//
#include <hip/hip_runtime.h>
#include <hip/hip_bf16.h>
#include <math.h>

// GraphFeatLearningLayer for MI455X (gfx1250, wave32, WMMA).
// Pipeline (all f32, using V_WMMA_F32_16X16X4_F32 for every GEMM):
//   k_alpha   : row-normalize alphas, scale by sqrt(64)=8
//   k_xbar    : X_bar = pc * a, row sq-norms g, zero colsum
//   k_gram    : G = X_bar X_bar^T (WMMA, K=64) -> Wu = thresh(exp(-(gn+gm-2G)/10))*mask
//               + column partial sums (LDS) -> global atomicAdd colsum
//   k_diffmul : out = Wu @ (U / colsum) + 0.5*U   (WMMA, K=1024 in 64-chunks) x8
//   k_absdiff : |U1-U2|
//   k_pool    : masked mean over nodes, 4 features x 64 dims per (b,w)
//
// Workspace (floats): 256 + 2*32768 + 7*2097152 + 32*1024*1024  (~190 MB)

namespace {

constexpr int kB = 8;
constexpr int kNW = 4;
constexpr int kN = 1024;
constexpr int kD = 64;
constexpr int kPairs = kB * kNW;          // 32
constexpr float kSigmaInv = 0.1f;         // 1/SIGMA
constexpr float kThresh = 1e-4f;
constexpr int kSlot = kPairs * kN * kD;   // floats per (pairs,N,D) matrix slot

typedef float v2f __attribute__((ext_vector_type(2)));
typedef float v8f __attribute__((ext_vector_type(8)));

__device__ inline v8f wmma4(v2f a, v2f b, v8f c) {
  // D = A(16x4 f32) * B(4x16 f32) + C(16x16 f32), wave32
  return __builtin_amdgcn_wmma_f32_16x16x4_f32(false, a, false, b, (short)0, c,
                                               false, false);
}

__device__ inline v8f zero8() {
  v8f z;
#pragma unroll
  for (int i = 0; i < 8; ++i) z[i] = 0.0f;
  return z;
}

// ---------------------------------------------------------------- k_alpha
__global__ __launch_bounds__(256) void k_alpha(const float* __restrict__ alphas,
                                               float* __restrict__ a_out) {
  __shared__ float red[256];
  const int tid = threadIdx.x;
  const int row = tid >> 6;   // 4 weight rows
  const int d = tid & 63;
  const float v = alphas[row * kD + d];
  red[tid] = v * v;
  __syncthreads();
  for (int s = 32; s >= 1; s >>= 1) {
    if ((tid & 63) < s) red[tid] += red[tid + s];
    __syncthreads();
  }
  const float norm = sqrtf(red[row << 6]);
  a_out[row * kD + d] = 8.0f * v / norm;  // sqrt(DIM)=8, row-normalized
}

// ---------------------------------------------------------------- k_xbar
// grid (pairs, N/128), 256 threads = 8 waves; each wave owns 16 rows.
__global__ __launch_bounds__(256) void k_xbar(const float* __restrict__ pc,
                                              const float* __restrict__ a_n,
                                              float* __restrict__ xbar,
                                              float* __restrict__ g,
                                              float* __restrict__ colsum) {
  const int p = blockIdx.x;
  const int nb = blockIdx.y * 128;
  const int b = p >> 2, w = p & 3;
  const int tid = threadIdx.x;
  const int wv = tid >> 5, lane = tid & 31;
  const float a0 = a_n[w * kD + lane];
  const float a1 = a_n[w * kD + lane + 32];
  for (int i = 0; i < 16; ++i) {
    const int n = nb + wv * 16 + i;
    const size_t src = ((size_t)b * kN + n) * kD + lane;
    const float x0 = pc[src] * a0;
    const float x1 = pc[src + 32] * a1;
    const size_t dst = ((size_t)p * kN + n) * kD + lane;
    xbar[dst] = x0;
    xbar[dst + 32] = x1;
    float s = x0 * x0 + x1 * x1;
    for (int off = 16; off >= 1; off >>= 1) s += __shfl_down(s, off, 32);
    if (lane == 0) g[p * kN + n] = s;
  }
  if (tid < 128) colsum[p * kN + nb + tid] = 0.0f;
}

// ---------------------------------------------------------------- k_gram
// grid (pairs, N/128 row tiles, N/64 col tiles), 256 threads = 8 waves.
// Each wave: 16 output rows x 64 cols = 4 f32-WMMA accumulators, K=64.
__global__ __launch_bounds__(256) void k_gram(
    const float* __restrict__ xbar, const float* __restrict__ g,
    const unsigned char* __restrict__ mask, float* __restrict__ wu,
    float* __restrict__ colsum) {
  __shared__ float At[128][68];  // pitch 68: row stride mod 64 == 4 -> no bank conflicts
  __shared__ float Bt[64][68];
  __shared__ float colp[64];
  const int p = blockIdx.x;
  const int nb = blockIdx.y * 128;
  const int mb = blockIdx.z * 64;
  const int tid = threadIdx.x;

  for (int idx = tid; idx < 128 * 64; idx += 256) {
    const int r = idx >> 6, c = idx & 63;
    At[r][c] = xbar[((size_t)p * kN + nb + r) * kD + c];
  }
  for (int idx = tid; idx < 64 * 64; idx += 256) {
    const int r = idx >> 6, c = idx & 63;
    Bt[r][c] = xbar[((size_t)p * kN + mb + r) * kD + c];
  }
  if (tid < 64) colp[tid] = 0.0f;
  __syncthreads();

  const int wv = tid >> 5, L = tid & 31;
  const int rowA = wv * 16 + (L & 15);
  const int hi2 = 2 * (L >> 4);  // A/B K-split across lane halves (ISA 16x4 layout)
  v8f acc[4];
#pragma unroll
  for (int c = 0; c < 4; ++c) acc[c] = zero8();

#pragma unroll
  for (int k0 = 0; k0 < 64; k0 += 4) {
    const int kk = k0 + hi2;
    v2f a;
    a.x = At[rowA][kk];
    a.y = At[rowA][kk + 1];
#pragma unroll
    for (int c = 0; c < 4; ++c) {
      v2f bb;  // B = X_bar^T : B[k][m] = Xbar[m][k]
      bb.x = Bt[c * 16 + (L & 15)][kk];
      bb.y = Bt[c * 16 + (L & 15)][kk + 1];
      acc[c] = wmma4(a, bb, acc[c]);
    }
  }

  // epilogue: C/D layout -> lane L, vgpr r holds (M = r + 8*(L>>4), Ncol = L&15)
  const int bidx = p >> 2;
  const unsigned char* mrow = mask + (size_t)bidx * kN;
  const int nBase = nb + wv * 16 + 8 * (L >> 4);
  float gn[8], mn[8];
#pragma unroll
  for (int r = 0; r < 8; ++r) {
    const int n = nBase + r;
    gn[r] = g[p * kN + n];
    mn[r] = mrow[n] ? 1.0f : 0.0f;
  }
#pragma unroll
  for (int c = 0; c < 4; ++c) {
    const int mcol = mb + c * 16 + (L & 15);
    const float gm = g[p * kN + mcol];
    const float mm = mrow[mcol] ? 1.0f : 0.0f;
    float csum = 0.0f;
#pragma unroll
    for (int r = 0; r < 8; ++r) {
      const int n = nBase + r;
      const float dd = gn[r] + gm - 2.0f * acc[c][r];
      float e = expf(-dd * kSigmaInv);
      e = (e < kThresh) ? 0.0f : e;
      e *= mn[r] * mm;
      wu[((size_t)p * kN + n) * kN + mcol] = e;
      csum += e;
    }
    atomicAdd(&colp[c * 16 + (L & 15)], csum);
  }
  __syncthreads();
  if (tid < 64) atomicAdd(&colsum[p * kN + mb + tid], colp[tid]);
}

// ---------------------------------------------------------------- k_diffmul
// out = Wu @ (U/colsum) + 0.5*U.  grid (pairs, N/128), 256 threads = 8 waves.
// K=1024 processed in 64-wide LDS chunks; Wu re-read comes from L2 (fits).
__global__ __launch_bounds__(256) void k_diffmul(
    const float* __restrict__ wu, const float* __restrict__ colsum,
    const float* __restrict__ uin, float* __restrict__ uout) {
  __shared__ float Wt[128][68];  // [out-row][k], A-style reads
  __shared__ float Bt[64][80];   // [k][d], B-style reads; pitch 80 (mod64==16)
  const int p = blockIdx.x;
  const int nb = blockIdx.y * 128;
  const int tid = threadIdx.x;
  const int wv = tid >> 5, L = tid & 31;
  const int rowA = wv * 16 + (L & 15);
  const int hi2 = 2 * (L >> 4);
  v8f acc[4];
#pragma unroll
  for (int c = 0; c < 4; ++c) acc[c] = zero8();

  for (int kc = 0; kc < kN; kc += 64) {
    __syncthreads();
    for (int idx = tid; idx < 128 * 64; idx += 256) {
      const int r = idx >> 6, c = idx & 63;
      Wt[r][c] = wu[((size_t)p * kN + nb + r) * kN + kc + c];
    }
    for (int idx = tid; idx < 64 * 64; idx += 256) {
      const int r = idx >> 6, c = idx & 63;
      const float cs = colsum[p * kN + kc + r];
      // fold the column normalization into the B operand
      Bt[r][c] = uin[((size_t)p * kN + kc + r) * kD + c] / fmaxf(cs, 1e-8f);
    }
    __syncthreads();
    if (kc + 64 < kN && tid < 128) {
      // pull next Wu chunk toward the caches (global_prefetch_b8)
      __builtin_prefetch(&wu[((size_t)p * kN + nb + tid) * kN + kc + 64], 0, 1);
    }
#pragma unroll
    for (int k0 = 0; k0 < 64; k0 += 4) {
      const int kk = k0 + hi2;
      v2f a;
      a.x = Wt[rowA][kk];
      a.y = Wt[rowA][kk + 1];
#pragma unroll
      for (int c = 0; c < 4; ++c) {
        v2f bb;
        bb.x = Bt[kk][c * 16 + (L & 15)];
        bb.y = Bt[kk + 1][c * 16 + (L & 15)];
        acc[c] = wmma4(a, bb, acc[c]);
      }
    }
  }

  const int nBase = nb + wv * 16 + 8 * (L >> 4);
#pragma unroll
  for (int c = 0; c < 4; ++c) {
    const int d = c * 16 + (L & 15);
#pragma unroll
    for (int r = 0; r < 8; ++r) {
      const size_t o = ((size_t)p * kN + nBase + r) * kD + d;
      uout[o] = acc[c][r] + 0.5f * uin[o];  // +0.5*I of the diffusion operator
    }
  }
}

// ---------------------------------------------------------------- k_absdiff
__global__ __launch_bounds__(256) void k_absdiff(const float* __restrict__ x,
                                                 const float* __restrict__ y,
                                                 float* __restrict__ dst, int n) {
  int i = blockIdx.x * blockDim.x + threadIdx.x;
  const int stride = gridDim.x * blockDim.x;
  for (; i < n; i += stride) dst[i] = fabsf(x[i] - y[i]);
}

// ---------------------------------------------------------------- k_pool
// grid (B, NW), 256 threads: thread = (feature f = tid>>6, dim d = tid&63).
// feats: [U4, |U1-U2| (=s5), |U2-U4|, |V2-V4|]
__global__ __launch_bounds__(256) void k_pool(
    const unsigned char* __restrict__ mask, const float* __restrict__ u2,
    const float* __restrict__ u4, const float* __restrict__ f1,
    const float* __restrict__ v2m, const float* __restrict__ v4m,
    float* __restrict__ out) {
  const int b = blockIdx.x, w = blockIdx.y;
  const int p = b * kNW + w;
  const int tid = threadIdx.x;
  const int f = tid >> 6, d = tid & 63;
  const size_t base = (size_t)p * kN * kD;
  const float* pu2 = u2 + base;
  const float* pu4 = u4 + base;
  const float* pf1 = f1 + base;
  const float* pv2 = v2m + base;
  const float* pv4 = v4m + base;
  float sum = 0.0f, cnt = 0.0f;
  for (int n = 0; n < kN; ++n) {
    const float mk = mask[(size_t)b * kN + n] ? 1.0f : 0.0f;
    cnt += mk;
    const size_t o = (size_t)n * kD + d;
    float val;
    if (f == 0)      val = pu4[o];
    else if (f == 1) val = pf1[o];
    else if (f == 2) val = fabsf(pu2[o] - pu4[o]);
    else             val = fabsf(pv2[o] - pv4[o]);
    sum += mk * val;
  }
  out[(size_t)b * (kNW * 4 * kD) + (size_t)w * (4 * kD) + f * kD + d] =
      sum / fmaxf(cnt, 1e-8f);
}

}  // namespace

extern "C" void kernel_launch(void* const* d_in, const int* in_sizes, int n_in,
                              void* d_out, int out_size, void* d_ws,
                              size_t ws_size, hipStream_t stream) {
  (void)in_sizes; (void)n_in; (void)out_size; (void)ws_size;
  const float* pc = (const float*)d_in[0];                  // (8,1024,64) f32
  const unsigned char* mask = (const unsigned char*)d_in[1];// (8,1024) bool
  const float* alphas = (const float*)d_in[2];              // (4,64) f32
  float* out = (float*)d_out;                               // (8,1024) f32

  float* wsf = (float*)d_ws;
  float* a_n = wsf; wsf += 256;
  float* g   = wsf; wsf += kPairs * kN;
  float* cs  = wsf; wsf += kPairs * kN;
  float* s0  = wsf; wsf += kSlot;   // X_bar (= U0)
  float* s1  = wsf; wsf += kSlot;
  float* s2  = wsf; wsf += kSlot;
  float* s3  = wsf; wsf += kSlot;
  float* s4  = wsf; wsf += kSlot;
  float* s5  = wsf; wsf += kSlot;
  float* s6  = wsf; wsf += kSlot;
  float* wuv = wsf;                 // (32,1024,1024) f32, un-normalized W

  const dim3 blk(256);
  k_alpha<<<1, blk, 0, stream>>>(alphas, a_n);
  k_xbar<<<dim3(kPairs, kN / 128), blk, 0, stream>>>(pc, a_n, s0, g, cs);
  k_gram<<<dim3(kPairs, kN / 128, kN / 64), blk, 0, stream>>>(s0, g, mask, wuv, cs);

  const dim3 dgrid(kPairs, kN / 128);
  k_diffmul<<<dgrid, blk, 0, stream>>>(wuv, cs, s0, s1);        // U1
  k_diffmul<<<dgrid, blk, 0, stream>>>(wuv, cs, s1, s2);        // U2
  k_diffmul<<<dgrid, blk, 0, stream>>>(wuv, cs, s2, s3);        // U3
  k_diffmul<<<dgrid, blk, 0, stream>>>(wuv, cs, s3, s4);        // U4
  k_absdiff<<<2048, blk, 0, stream>>>(s1, s2, s5, kSlot);       // f1 = |U1-U2| = V0
  k_diffmul<<<dgrid, blk, 0, stream>>>(wuv, cs, s5, s6);        // V1
  k_diffmul<<<dgrid, blk, 0, stream>>>(wuv, cs, s6, s1);        // V2 -> s1
  k_diffmul<<<dgrid, blk, 0, stream>>>(wuv, cs, s1, s6);        // V3 -> s6
  k_diffmul<<<dgrid, blk, 0, stream>>>(wuv, cs, s6, s3);        // V4 -> s3

  k_pool<<<dim3(kB, kNW), blk, 0, stream>>>(mask, s2, s4, s5, s1, s3, out);
}